// Tree_INIT_54623394070803
// MI455X (gfx1250) — compile-verified
//
#include <hip/hip_runtime.h>
#include <hip/hip_bf16.h>

// ---------------- problem constants (fixed by setup_inputs) ----------------
constexpr int N_NODES = 50000;
constexpr int N_EDGES = 150000;
constexpr int N_RINGS = 30000;
constexpr int NNZ     = 300000;
constexpr int H       = 64;   // hidden width
constexpr int FDN     = 16;
constexpr int FDE     = 8;
constexpr int L       = 4;    // levels
constexpr int S1 = 100000, S2 = 200000, S3 = 400000;
constexpr float BN_EPS = 1e-5f;

// ---------------- output layout (floats), return order (tn, te, tr) -------
constexpr size_t O_TN0 = 0;
constexpr size_t O_TN1 = O_TN0 + (size_t)N_NODES * H;
constexpr size_t O_TN2 = O_TN1 + (size_t)S1 * H;
constexpr size_t O_TN3 = O_TN2 + (size_t)S2 * H;
constexpr size_t O_TE0 = O_TN3 + (size_t)S3 * H;
constexpr size_t O_TE1 = O_TE0 + (size_t)N_EDGES * H;
constexpr size_t O_TE2 = O_TE1 + (size_t)S1 * H;
constexpr size_t O_TE3 = O_TE2 + (size_t)S2 * H;
constexpr size_t O_TR0 = O_TE3 + (size_t)S3 * H;
constexpr size_t O_TR1 = O_TR0 + (size_t)N_RINGS * H;
constexpr size_t O_TR2 = O_TR1 + (size_t)S1 * H;
constexpr size_t O_TR3 = O_TR2 + (size_t)S2 * H;

// ---------------- workspace layout (floats) --------------------------------
constexpr size_t W_NF    = 0;                                     // [L,N,64]
constexpr size_t W_EF    = W_NF   + (size_t)L * N_NODES * H;      // [L,E,64]
constexpr size_t W_RACC  = W_EF   + (size_t)L * N_EDGES * H;      // [L,R,64]
constexpr size_t W_RPRE  = W_RACC + (size_t)L * N_RINGS * H;      // [L,R,64]
constexpr size_t W_R2E   = W_RPRE + (size_t)L * N_RINGS * H;      // [3,E,64]
constexpr size_t W_STATS = W_R2E  + (size_t)3 * N_EDGES * H;      // 3*1024
// per-entity stat block (1024 floats): sum[256], sumsq[256], scale[256], shift[256]

typedef __attribute__((ext_vector_type(2))) float v2f;
typedef __attribute__((ext_vector_type(8))) float v8f;

// ---------------------------------------------------------------------------
// Fused GEMM (Y = X @ W[l] + b[l]) + per-column sum / sum-of-squares.
// One wave handles a 16-row tile across all 64 output columns using
// V_WMMA_F32_16X16X4_F32: 4 N-tiles x (K/4) K-chunks. M % 16 == 0 always.
// A layout (ISA 7.12.2, 32-bit 16x4): lanes 0-15 -> K=0,1 ; lanes 16-31 -> K=2,3
// B layout (4x16, like C with K rows): vgpr v, K = v + 2*(lane>=16)
// C layout: vgpr v, row M = v + 8*(lane>=16), col N = lane%16
// ---------------------------------------------------------------------------
template <int K>
__global__ void gemm_bias_stats(const float* __restrict__ X, size_t x_lvl_stride,
                                const float* __restrict__ W,      // [L,K,64]
                                const float* __restrict__ bias,   // [L,64]
                                float* __restrict__ Y,            // [L,M,64]
                                float* __restrict__ stat_sum,     // [L,64]
                                float* __restrict__ stat_sq,      // [L,64]
                                int M) {
  const int lane  = threadIdx.x & 31;
  const int wave  = threadIdx.x >> 5;
  const int tile  = blockIdx.x * (blockDim.x >> 5) + wave;
  const int lvl   = blockIdx.y;
  const int mtile = M >> 4;
  if (tile >= mtile) return;                 // wave-uniform exit (EXEC stays full)

  const int row0 = tile << 4;
  const int hi   = lane >> 4;                // 0: K lo pair, 1: K hi pair
  const int mrow = row0 + (lane & 15);

  const float* Xl = X + x_lvl_stride * (size_t)lvl;

  v2f a[K / 4];
#pragma unroll
  for (int kc = 0; kc < K / 4; ++kc) {
    a[kc] = *(const v2f*)(Xl + (size_t)mrow * K + kc * 4 + hi * 2);
  }

  const float* Wl = W + (size_t)lvl * K * H;
  const float* bl = bias + (size_t)lvl * H;
  float* Yl       = Y + (size_t)lvl * M * H;

#pragma unroll
  for (int nt = 0; nt < 4; ++nt) {
    const int col = nt * 16 + (lane & 15);
    v8f c = {};
#pragma unroll
    for (int kc = 0; kc < K / 4; ++kc) {
      v2f b;
      b.x = Wl[(kc * 4 + hi * 2 + 0) * H + col];
      b.y = Wl[(kc * 4 + hi * 2 + 1) * H + col];
      c = __builtin_amdgcn_wmma_f32_16x16x4_f32(false, a[kc], false, b,
                                                (short)0, c, false, false);
    }
    const float bv = bl[col];
    float s = 0.f, s2 = 0.f;
#pragma unroll
    for (int v = 0; v < 8; ++v) {
      float y = c[v] + bv;
      Yl[(size_t)(row0 + v + hi * 8) * H + col] = y;
      s += y;
      s2 += y * y;
    }
    // fold the two half-wave partners (same column, rows 0-7 vs 8-15)
    s  += __shfl_xor(s, 16, 32);
    s2 += __shfl_xor(s2, 16, 32);
    if (hi == 0) {
      atomicAdd(&stat_sum[lvl * H + col], s);
      atomicAdd(&stat_sq[lvl * H + col], s2);
    }
  }
}

// Fold BN into per-column scale/shift: y_norm = y*scale + shift, then ReLU.
__global__ void finalize_stats(const float* __restrict__ stat_sum,
                               const float* __restrict__ stat_sq,
                               const float* __restrict__ gamma,
                               const float* __restrict__ beta,
                               float* __restrict__ scale,
                               float* __restrict__ shift, int M) {
  int i = blockIdx.x * blockDim.x + threadIdx.x;
  if (i >= L * H) return;
  float inv_n = 1.0f / (float)M;
  float mean  = stat_sum[i] * inv_n;
  float var   = stat_sq[i] * inv_n - mean * mean;
  float is    = rsqrtf(var + BN_EPS);
  float sc    = gamma[i] * is;
  scale[i]    = sc;
  shift[i]    = beta[i] - mean * sc;
}

// In-place BN+ReLU over [L,M,64]; level 0 additionally streamed to d_out.
__global__ void bn_relu_apply(float* __restrict__ Y,
                              const float* __restrict__ scale,
                              const float* __restrict__ shift, int M,
                              float* __restrict__ out0) {
  const int lvl = blockIdx.y;
  const size_t n4 = (size_t)M * (H / 4);
  size_t i = (size_t)blockIdx.x * blockDim.x + threadIdx.x;
  if (i >= n4) return;
  float4* Yl = (float4*)(Y + (size_t)lvl * M * H);
  const float* sc = scale + lvl * H;
  const float* sh = shift + lvl * H;
  const int c0 = (int)((i * 4) & (H - 1));
  float4 y = Yl[i];
  y.x = fmaxf(0.f, y.x * sc[c0 + 0] + sh[c0 + 0]);
  y.y = fmaxf(0.f, y.y * sc[c0 + 1] + sh[c0 + 1]);
  y.z = fmaxf(0.f, y.z * sc[c0 + 2] + sh[c0 + 2]);
  y.w = fmaxf(0.f, y.w * sc[c0 + 3] + sh[c0 + 3]);
  Yl[i] = y;
  if (lvl == 0 && out0 != nullptr) ((float4*)out0)[i] = y;
}

// dst[i,:] = src[idx[i],:]  (128-bit vectorized)
__global__ void gather_rows(const float* __restrict__ src,
                            const int* __restrict__ idx,
                            float* __restrict__ dst, int n) {
  size_t t = (size_t)blockIdx.x * blockDim.x + threadIdx.x;
  if (t >= (size_t)n * (H / 4)) return;
  const int i  = (int)(t >> 4);
  const int c4 = (int)(t & 15);
  const int r  = idx[i];
  ((float4*)dst)[t] = ((const float4*)src)[(size_t)r * (H / 4) + c4];
}

// dst[didx[j],:] += src[sidx[j],:]  (f32 global atomics, coalesced by column)
__global__ void scatter_add(float* __restrict__ dst, size_t dst_lvl_stride,
                            const float* __restrict__ src, size_t src_lvl_stride,
                            const int* __restrict__ didx,
                            const int* __restrict__ sidx, int nnz) {
  const int lvl = blockIdx.y;
  size_t t = (size_t)blockIdx.x * blockDim.x + threadIdx.x;
  if (t >= (size_t)nnz * H) return;
  const int j = (int)(t >> 6);
  const int c = (int)(t & 63);
  float v = src[src_lvl_stride * lvl + (size_t)sidx[j] * H + c];
  atomicAdd(&dst[dst_lvl_stride * lvl + (size_t)didx[j] * H + c], v);
}

// ---------------------------------------------------------------------------
extern "C" void kernel_launch(void* const* d_in, const int* in_sizes, int n_in,
                              void* d_out, int out_size, void* d_ws, size_t ws_size,
                              hipStream_t stream) {
  const float* node_attr = (const float*)d_in[0];
  const float* edge_attr = (const float*)d_in[1];
  const int* n_img[3] = {(const int*)d_in[2], (const int*)d_in[3], (const int*)d_in[4]};
  const int* e_img[3] = {(const int*)d_in[5], (const int*)d_in[6], (const int*)d_in[7]};
  const int* ring_idx = (const int*)d_in[8];
  const int* edge_idx = (const int*)d_in[9];
  const float* Wn = (const float*)d_in[12];
  const float* bn_b = (const float*)d_in[13];
  const float* n_gamma = (const float*)d_in[14];
  const float* n_beta = (const float*)d_in[15];
  const float* We = (const float*)d_in[16];
  const float* be = (const float*)d_in[17];
  const float* e_gamma = (const float*)d_in[18];
  const float* e_beta = (const float*)d_in[19];
  const float* Wr = (const float*)d_in[20];
  const float* br = (const float*)d_in[21];
  const float* r_gamma = (const float*)d_in[22];
  const float* r_beta = (const float*)d_in[23];

  float* ws  = (float*)d_ws;
  float* out = (float*)d_out;

  float* nf   = ws + W_NF;
  float* ef   = ws + W_EF;
  float* racc = ws + W_RACC;
  float* rpre = ws + W_RPRE;
  float* r2e  = ws + W_R2E;
  float* st_n = ws + W_STATS;            // [sum|sumsq|scale|shift] x 256
  float* st_e = st_n + 1024;
  float* st_r = st_e + 1024;

  // ---- zero accumulators & stats (capture-safe memset nodes) ----
  hipMemsetAsync(racc, 0, (size_t)L * N_RINGS * H * sizeof(float), stream);
  hipMemsetAsync(r2e, 0, (size_t)3 * N_EDGES * H * sizeof(float), stream);
  hipMemsetAsync(st_n, 0, 3 * 1024 * sizeof(float), stream);

  const int TB = 256;
  const int wavesPerBlock = TB / 32;

  // ---- node / edge encoders: GEMM + bias + BN stats ----
  {
    dim3 g((N_NODES / 16 + wavesPerBlock - 1) / wavesPerBlock, L);
    gemm_bias_stats<FDN><<<g, TB, 0, stream>>>(node_attr, 0, Wn, bn_b, nf,
                                               st_n + 0, st_n + 256, N_NODES);
  }
  {
    dim3 g((N_EDGES / 16 + wavesPerBlock - 1) / wavesPerBlock, L);
    gemm_bias_stats<FDE><<<g, TB, 0, stream>>>(edge_attr, 0, We, be, ef,
                                               st_e + 0, st_e + 256, N_EDGES);
  }
  finalize_stats<<<1, 256, 0, stream>>>(st_n, st_n + 256, n_gamma, n_beta,
                                        st_n + 512, st_n + 768, N_NODES);
  finalize_stats<<<1, 256, 0, stream>>>(st_e, st_e + 256, e_gamma, e_beta,
                                        st_e + 512, st_e + 768, N_EDGES);

  // ---- BN+ReLU in place; level 0 copies directly into d_out ----
  {
    dim3 g(((size_t)N_NODES * 16 + TB - 1) / TB, L);
    bn_relu_apply<<<g, TB, 0, stream>>>(nf, st_n + 512, st_n + 768, N_NODES,
                                        out + O_TN0);
  }
  {
    dim3 g(((size_t)N_EDGES * 16 + TB - 1) / TB, L);
    bn_relu_apply<<<g, TB, 0, stream>>>(ef, st_e + 512, st_e + 768, N_EDGES,
                                        out + O_TE0);
  }

  // ---- tn / te gathers (levels 1..3) ----
  const int nsz[3] = {S1, S2, S3};
  const size_t tn_off[3] = {O_TN1, O_TN2, O_TN3};
  const size_t te_off[3] = {O_TE1, O_TE2, O_TE3};
  for (int k = 1; k < L; ++k) {
    int n = nsz[k - 1];
    size_t nt = (size_t)n * 16;
    gather_rows<<<(nt + TB - 1) / TB, TB, 0, stream>>>(
        nf + (size_t)k * N_NODES * H, n_img[k - 1], out + tn_off[k - 1], n);
    gather_rows<<<(nt + TB - 1) / TB, TB, 0, stream>>>(
        ef + (size_t)k * N_EDGES * H, e_img[k - 1], out + te_off[k - 1], n);
  }

  // ---- edge -> ring segment sum (all 4 levels) ----
  {
    dim3 g(((size_t)NNZ * H + TB - 1) / TB, L);
    scatter_add<<<g, TB, 0, stream>>>(racc, (size_t)N_RINGS * H,
                                      ef, (size_t)N_EDGES * H,
                                      ring_idx, edge_idx, NNZ);
  }

  // ---- ring encoder: GEMM(K=64) + BN ----
  {
    dim3 g((N_RINGS / 16 + wavesPerBlock - 1) / wavesPerBlock, L);
    gemm_bias_stats<H><<<g, TB, 0, stream>>>(racc, (size_t)N_RINGS * H, Wr, br,
                                             rpre, st_r + 0, st_r + 256, N_RINGS);
  }
  finalize_stats<<<1, 256, 0, stream>>>(st_r, st_r + 256, r_gamma, r_beta,
                                        st_r + 512, st_r + 768, N_RINGS);
  {
    dim3 g(((size_t)N_RINGS * 16 + TB - 1) / TB, L);
    bn_relu_apply<<<g, TB, 0, stream>>>(rpre, st_r + 512, st_r + 768, N_RINGS,
                                        out + O_TR0);
  }

  // ---- ring -> edge segment sum for levels 1..3 ----
  {
    dim3 g(((size_t)NNZ * H + TB - 1) / TB, 3);
    scatter_add<<<g, TB, 0, stream>>>(r2e, (size_t)N_EDGES * H,
                                      rpre + (size_t)N_RINGS * H,  // level 1 base
                                      (size_t)N_RINGS * H,
                                      edge_idx, ring_idx, NNZ);
  }

  // ---- tr gathers (levels 1..3) ----
  const size_t tr_off[3] = {O_TR1, O_TR2, O_TR3};
  for (int k = 1; k < L; ++k) {
    int n = nsz[k - 1];
    size_t nt = (size_t)n * 16;
    gather_rows<<<(nt + TB - 1) / TB, TB, 0, stream>>>(
        r2e + (size_t)(k - 1) * N_EDGES * H, e_img[k - 1], out + tr_off[k - 1], n);
  }
}